// RFFT_86870008529604
// MI455X (gfx1250) — compile-verified
//
#include <hip/hip_runtime.h>
#include <hip/hip_bf16.h>
#include <math.h>

// Four-step FFT (8192 = 64 x 128) as fp32 WMMA matmuls on gfx1250.
//   Stage1: per (b, n1): Y[k2][c] = sum_n2 x[n2*64+n1][c] * W128^(n2*k2),
//           Z = Y * W8192^(n1*k2)          (128x128 DFT) x (128x32) matmul
//   Stage2: per (b, k2): X[k1][c] = sum_n1 Z[n1][k2][c] * W64^(n1*k1), k1<32
//           out = |X|                       (32x64) x (64x32) complex matmul
// Channels (C=32) map onto two WMMA N=16 tiles -> contiguous B-operand loads.
// Z is stored as interleaved float2 (re,im) for wide, coalesced VMEM.
// Stage-1 input tile is staged to LDS with global_load_async_to_lds_b128
// (ASYNCcnt path), then consumed as WMMA B-fragments via ds loads.
//
// Workspace layout (floats):
//   [0      ..16384) F128 cos   [16384..32768) F128 -sin
//   [32768  ..40960) TW   cos   [40960..49152) TW   -sin   (64 x 128)
//   [49152  ..51200) F64  cos   [51200..53248) F64  -sin   (32 x 64)
//   [65536  ..)      Z interleaved float2, 128*8192*32 elements (256 MB)

typedef __attribute__((ext_vector_type(2))) float v2f;
typedef __attribute__((ext_vector_type(8))) float v8f;

#define WMMA_F32X4(A, Bf, Cc) \
    __builtin_amdgcn_wmma_f32_16x16x4_f32(false, (A), false, (Bf), (short)0, (Cc), false, false)

#define WS_F128C 0
#define WS_F128S 16384
#define WS_TWC   32768
#define WS_TWS   40960
#define WS_F64C  49152
#define WS_F64S  51200
#define WS_Z     65536      // float index; float2 array of 128*8192*32 elems

__global__ void fft_tables(float* __restrict__ tab) {
    const int i = blockIdx.x * 256 + threadIdx.x;       // 0..16383
    const float PI2 = 6.2831853071795864769f;
    {
        int k = i >> 7, n = i & 127;
        float th = PI2 * (float)((k * n) & 127) * (1.0f / 128.0f);
        tab[WS_F128C + i] = cosf(th);
        tab[WS_F128S + i] = -sinf(th);
    }
    if (i < 8192) {
        int n1 = i >> 7, k2 = i & 127;                  // n1*k2 < 8192
        float th = PI2 * (float)(n1 * k2) * (1.0f / 8192.0f);
        tab[WS_TWC + i] = cosf(th);
        tab[WS_TWS + i] = -sinf(th);
    }
    if (i < 2048) {
        int k1 = i >> 6, n1 = i & 63;
        float th = PI2 * (float)((k1 * n1) & 63) * (1.0f / 64.0f);
        tab[WS_F64C + i] = cosf(th);
        tab[WS_F64S + i] = -sinf(th);
    }
}

// grid: (n1=64, b=128), block 256 (8 waves; wave w owns M-tile w of k2)
__global__ void __launch_bounds__(256)
fft_stage1(const float* __restrict__ x, const float* __restrict__ tab,
           float2* __restrict__ zri) {
    const int n1 = blockIdx.x;
    const int b  = blockIdx.y;
    const float* F128C = tab + WS_F128C;
    const float* F128S = tab + WS_F128S;
    const float* TWC   = tab + WS_TWC;
    const float* TWS   = tab + WS_TWS;

    __shared__ float sX[128 * 32];   // Xsub[n2][c] = x[b][n2*64+n1][c]
    const int t = threadIdx.x;

    // Async DMA: global -> LDS, 16 bytes per lane per op, no VGPR data path.
    for (int f = t; f < 1024; f += 256) {               // 1024 x 16B = 16 KB
        const int n2 = f >> 3;
        const int cp = (f & 7) << 2;
        const float* gsrc =
            &x[(((size_t)b * 8192) + (size_t)n2 * 64 + n1) * 32 + cp];
        const unsigned ldst =
            (unsigned)(unsigned long long)&sX[n2 * 32 + cp];
        asm volatile("global_load_async_to_lds_b128 %0, %1, off"
                     :: "v"(ldst), "v"(gsrc) : "memory");
    }
    asm volatile("s_wait_asynccnt 0x0" ::: "memory");
    __syncthreads();

    const int wave = t >> 5;          // M tile (k2 block of 16)
    const int lane = t & 31;
    const int m16  = lane & 15;
    const int half = lane >> 4;
    const int k2row = wave * 16 + m16;

    v8f accRe0 = {}, accIm0 = {}, accRe1 = {}, accIm1 = {};
    const float* arC = &F128C[k2row * 128];
    const float* arS = &F128S[k2row * 128];

#pragma unroll 4
    for (int ks = 0; ks < 128; ks += 4) {
        const int kr = ks + 2 * half;
        v2f ac = *(const v2f*)&arC[kr];
        v2f as = *(const v2f*)&arS[kr];
        v2f b0, b1;
        b0.x = sX[(kr + 0) * 32 + m16];
        b0.y = sX[(kr + 1) * 32 + m16];
        b1.x = sX[(kr + 0) * 32 + 16 + m16];
        b1.y = sX[(kr + 1) * 32 + 16 + m16];
        accRe0 = WMMA_F32X4(ac, b0, accRe0);
        accIm0 = WMMA_F32X4(as, b0, accIm0);
        accRe1 = WMMA_F32X4(ac, b1, accRe1);
        accIm1 = WMMA_F32X4(as, b1, accIm1);
    }

    // twiddle by W8192^(n1*k2) and store Z as interleaved (re,im)
#pragma unroll
    for (int r = 0; r < 8; ++r) {
        const int k2 = wave * 16 + (half ? r + 8 : r);
        const float tc = TWC[n1 * 128 + k2];
        const float ts = TWS[n1 * 128 + k2];
        const size_t base = (((size_t)b * 64 + n1) * 128 + k2) * 32;
        float2 z;
        float yre = accRe0[r], yim = accIm0[r];
        z.x = yre * tc - yim * ts;
        z.y = yre * ts + yim * tc;
        zri[base + m16] = z;
        yre = accRe1[r]; yim = accIm1[r];
        z.x = yre * tc - yim * ts;
        z.y = yre * ts + yim * tc;
        zri[base + 16 + m16] = z;
    }
}

// grid: (k2g=32, b=128), block 256; wave w -> k2 = k2g*4 + w/2, M-tile = w&1
__global__ void __launch_bounds__(256)
fft_stage2(const float* __restrict__ tab, const float2* __restrict__ zri,
           float* __restrict__ out) {
    const int k2g = blockIdx.x;
    const int b   = blockIdx.y;
    const float* F64C = tab + WS_F64C;
    const float* F64S = tab + WS_F64S;

    const int t    = threadIdx.x;
    const int wave = t >> 5;
    const int lane = t & 31;
    const int m16  = lane & 15;
    const int half = lane >> 4;
    const int k2   = k2g * 4 + (wave >> 1);
    const int mt   = wave & 1;
    const int k1row = mt * 16 + m16;                    // 0..31

    v8f p1 = {}, p2 = {}, p3 = {}, p4 = {};
    v8f q1 = {}, q2 = {}, q3 = {}, q4 = {};
    const float* arC = &F64C[k1row * 64];
    const float* arS = &F64S[k1row * 64];

#pragma unroll 4
    for (int ks = 0; ks < 64; ks += 4) {
        const int n1a = ks + 2 * half;
        v2f ac = *(const v2f*)&arC[n1a];
        v2f as = *(const v2f*)&arS[n1a];
        const size_t zb0 = (((size_t)b * 64 + n1a) * 128 + k2) * 32;
        const size_t zb1 = zb0 + (size_t)128 * 32;      // n1a+1
        // prefetch next K-step's lines toward the WGP
        __builtin_prefetch(&zri[zb0 + (size_t)4 * 128 * 32 + m16], 0, 0);
        const float2 z00 = zri[zb0 + m16];
        const float2 z01 = zri[zb1 + m16];
        const float2 z10 = zri[zb0 + 16 + m16];
        const float2 z11 = zri[zb1 + 16 + m16];
        v2f br0, bi0, br1, bi1;
        br0.x = z00.x; br0.y = z01.x;
        bi0.x = z00.y; bi0.y = z01.y;
        br1.x = z10.x; br1.y = z11.x;
        bi1.x = z10.y; bi1.y = z11.y;
        p1 = WMMA_F32X4(ac, br0, p1);   // Fc*Zre
        p2 = WMMA_F32X4(as, bi0, p2);   // Fs*Zim
        p3 = WMMA_F32X4(ac, bi0, p3);   // Fc*Zim
        p4 = WMMA_F32X4(as, br0, p4);   // Fs*Zre
        q1 = WMMA_F32X4(ac, br1, q1);
        q2 = WMMA_F32X4(as, bi1, q2);
        q3 = WMMA_F32X4(ac, bi1, q3);
        q4 = WMMA_F32X4(as, br1, q4);
    }

#pragma unroll
    for (int r = 0; r < 8; ++r) {
        const int k1 = mt * 16 + (half ? r + 8 : r);
        const size_t ob = ((size_t)b * 4096 + (size_t)k1 * 128 + k2) * 32;
        float xre = p1[r] - p2[r];
        float xim = p3[r] + p4[r];
        out[ob + m16] = sqrtf(xre * xre + xim * xim);
        xre = q1[r] - q2[r];
        xim = q3[r] + q4[r];
        out[ob + 16 + m16] = sqrtf(xre * xre + xim * xim);
    }
}

extern "C" void kernel_launch(void* const* d_in, const int* in_sizes, int n_in,
                              void* d_out, int out_size, void* d_ws, size_t ws_size,
                              hipStream_t stream) {
    const float* x = (const float*)d_in[0];
    float* out = (float*)d_out;
    float* ws  = (float*)d_ws;
    float2* zri = (float2*)(ws + WS_Z);

    fft_tables<<<64, 256, 0, stream>>>(ws);
    fft_stage1<<<dim3(64, 128), 256, 0, stream>>>(x, ws, zri);
    fft_stage2<<<dim3(32, 128), 256, 0, stream>>>(ws, zri, out);
}